// DropBlock_54271206752261
// MI455X (gfx1250) — compile-verified
//
#include <hip/hip_runtime.h>

// 128-bit vector types
typedef float __attribute__((ext_vector_type(4))) f32x4;
typedef int   __attribute__((ext_vector_type(4))) i32x4;

#define AS1 __attribute__((address_space(1)))
#define AS3 __attribute__((address_space(3)))

#ifndef __has_builtin
#define __has_builtin(x) 0
#endif

#if __has_builtin(__builtin_amdgcn_global_load_async_to_lds_b128)
#define HAVE_ASYNC_LDS 1
#else
#define HAVE_ASYNC_LDS 0
#endif

static constexpr int kC       = 256;
static constexpr int kH       = 64;
static constexpr int kW       = 64;
static constexpr int kHW      = kH * kW;       // 4096 floats = 16 KB
static constexpr int kN       = 32;
static constexpr int kNPerBlk = 4;             // batches per apply-block
static constexpr int kG       = kN / kNPerBlk; // 8 batch-groups

// gamma computed in double (as in the Python reference) then rounded to f32,
// matching JAX's weak-typed scalar comparison against an f32 array.
__device__ constexpr float kGamma =
    (float)((1.0 - 0.9) / (7.0 * 7.0) * (64.0 * 64.0) / (58.0 * 58.0));

// Stage one 16 KB float plane into LDS via gfx1250 async DMA (4x b128/thread).
__device__ __forceinline__ void stage_plane_to_lds(const float* gsrc, float* lds,
                                                   int t) {
#if HAVE_ASYNC_LDS
  #pragma unroll
  for (int k = 0; k < 4; ++k) {
    const int i = (t + k * 256) * 4; // float index, 16-byte granules
    __builtin_amdgcn_global_load_async_to_lds_b128(
        (AS1 i32x4*)(gsrc + i), (AS3 i32x4*)(lds + i),
        /*offset=*/0, /*cpol=*/0);
  }
#if __has_builtin(__builtin_amdgcn_s_wait_asynccnt)
  __builtin_amdgcn_s_wait_asynccnt(0);
#else
  asm volatile("s_wait_asynccnt 0" ::: "memory");
#endif
#else
  const f32x4* s4 = (const f32x4*)gsrc;
  f32x4* d4 = (f32x4*)lds;
  #pragma unroll
  for (int k = 0; k < 4; ++k) {
    const int i = t + k * 256;
    d4[i] = s4[i];
  }
#endif
}

// Separable 7x7 max dilate of bordered Bernoulli centers; input plane in
// s_in (noise), scratch s_tmp. Result mask value per pixel emitted via
// direct call-side store (returned through lambda-free loop below).
// ---------------------------------------------------------------------------
// Kernel 1: per-channel mask -> d_ws (4 MB, stays L2-resident; 192 MB L2).
__global__ __launch_bounds__(256)
void dropblock_mask(const float* __restrict__ noise, float* __restrict__ mask) {
  __shared__ float s_noise[kHW];
  __shared__ float s_tmp[kHW];

  const int c = blockIdx.x;
  const int t = threadIdx.x;

  stage_plane_to_lds(noise + (size_t)c * kHW, s_noise, t);
  __syncthreads();

  // Pass 1: horizontal dilate. Center validity: row in [3,60], col in [3,60].
  #pragma unroll
  for (int k = 0; k < 16; ++k) {
    const int px = t + k * 256;
    const int h  = px >> 6;
    const int w  = px & 63;
    float hm = 0.0f;
    if (h >= 3 && h <= 60) {
      const int w0 = (w - 3 < 3)  ? 3  : (w - 3);
      const int w1 = (w + 3 > 60) ? 60 : (w + 3);
      const float* row = s_noise + (h << 6);
      for (int ww = w0; ww <= w1; ++ww)
        hm = fmaxf(hm, (row[ww] < kGamma) ? 1.0f : 0.0f);
    }
    s_tmp[px] = hm;
  }
  __syncthreads();

  // Pass 2: vertical dilate, write final mask (1 - dilated) to global (RT).
  float* mplane = mask + (size_t)c * kHW;
  #pragma unroll
  for (int k = 0; k < 16; ++k) {
    const int px = t + k * 256;
    const int h  = px >> 6;
    const int w  = px & 63;
    const int h0 = (h - 3 < 0)  ? 0  : (h - 3);
    const int h1 = (h + 3 > 63) ? 63 : (h + 3);
    float vm = 0.0f;
    for (int hh = h0; hh <= h1; ++hh)
      vm = fmaxf(vm, s_tmp[(hh << 6) + w]);
    mplane[px] = 1.0f - vm;
  }
}

// ---------------------------------------------------------------------------
// Kernel 2: streaming broadcast multiply. Mask plane async-DMA'd into LDS,
// hoisted to registers, then 4 batch planes streamed with NT b128 traffic.
__global__ __launch_bounds__(256)
void dropblock_apply(const float* __restrict__ x, const float* __restrict__ mask,
                     float* __restrict__ out) {
  __shared__ float s_mask[kHW];

  const int c = blockIdx.x;
  const int g = blockIdx.y;
  const int t = threadIdx.x;

  stage_plane_to_lds(mask + (size_t)c * kHW, s_mask, t);
  __syncthreads();

  // Hoist this thread's 16 mask values into registers (4x ds_load_b128).
  const f32x4* m4 = (const f32x4*)s_mask;
  f32x4 mv[4];
  #pragma unroll
  for (int k = 0; k < 4; ++k)
    mv[k] = m4[t + k * 256];

  const int n0 = g * kNPerBlk;
  #pragma unroll
  for (int nn = 0; nn < kNPerBlk; ++nn) {
    const size_t plane = ((size_t)(n0 + nn) * kC + c) * (size_t)kHW;
    const f32x4* xp = (const f32x4*)(x + plane);
    f32x4*       op = (f32x4*)(out + plane);
    #pragma unroll
    for (int k = 0; k < 4; ++k) {
      const int i = t + k * 256;
      f32x4 xv = __builtin_nontemporal_load(xp + i); // stream-once: NT load
      __builtin_nontemporal_store(xv * mv[k], op + i); // stream-once: NT store
    }
  }
}

// ---------------------------------------------------------------------------
// Fallback: fully fused single kernel (used only if d_ws is too small).
__global__ __launch_bounds__(256)
void dropblock_fused(const float* __restrict__ x,
                     const float* __restrict__ noise,
                     float* __restrict__ out) {
  __shared__ float s_mask[kHW];
  __shared__ float s_tmp[kHW];

  const int c = blockIdx.x;
  const int g = blockIdx.y;
  const int t = threadIdx.x;

  stage_plane_to_lds(noise + (size_t)c * kHW, s_mask, t);
  __syncthreads();

  #pragma unroll
  for (int k = 0; k < 16; ++k) {
    const int px = t + k * 256;
    const int h  = px >> 6;
    const int w  = px & 63;
    float hm = 0.0f;
    if (h >= 3 && h <= 60) {
      const int w0 = (w - 3 < 3)  ? 3  : (w - 3);
      const int w1 = (w + 3 > 60) ? 60 : (w + 3);
      const float* row = s_mask + (h << 6);
      for (int ww = w0; ww <= w1; ++ww)
        hm = fmaxf(hm, (row[ww] < kGamma) ? 1.0f : 0.0f);
    }
    s_tmp[px] = hm;
  }
  __syncthreads();

  #pragma unroll
  for (int k = 0; k < 16; ++k) {
    const int px = t + k * 256;
    const int h  = px >> 6;
    const int w  = px & 63;
    const int h0 = (h - 3 < 0)  ? 0  : (h - 3);
    const int h1 = (h + 3 > 63) ? 63 : (h + 3);
    float vm = 0.0f;
    for (int hh = h0; hh <= h1; ++hh)
      vm = fmaxf(vm, s_tmp[(hh << 6) + w]);
    s_mask[px] = 1.0f - vm;
  }
  __syncthreads();

  const f32x4* m4 = (const f32x4*)s_mask;
  f32x4 mv[4];
  #pragma unroll
  for (int k = 0; k < 4; ++k)
    mv[k] = m4[t + k * 256];

  const int n0 = g * kNPerBlk;
  #pragma unroll
  for (int nn = 0; nn < kNPerBlk; ++nn) {
    const size_t plane = ((size_t)(n0 + nn) * kC + c) * (size_t)kHW;
    const f32x4* xp = (const f32x4*)(x + plane);
    f32x4*       op = (f32x4*)(out + plane);
    #pragma unroll
    for (int k = 0; k < 4; ++k) {
      const int i = t + k * 256;
      f32x4 xv = __builtin_nontemporal_load(xp + i);
      __builtin_nontemporal_store(xv * mv[k], op + i);
    }
  }
}

extern "C" void kernel_launch(void* const* d_in, const int* in_sizes, int n_in,
                              void* d_out, int out_size, void* d_ws, size_t ws_size,
                              hipStream_t stream) {
  const float* x     = (const float*)d_in[0]; // [32, 256, 64, 64] f32
  const float* noise = (const float*)d_in[1]; // [256, 64, 64] f32
  float* out = (float*)d_out;                 // [32, 256, 64, 64] f32

  const size_t mask_bytes = (size_t)kC * kHW * sizeof(float); // 4 MB

  if (d_ws != nullptr && ws_size >= mask_bytes) {
    float* mask = (float*)d_ws;
    dropblock_mask<<<dim3(kC), 256, 0, stream>>>(noise, mask);
    dropblock_apply<<<dim3(kC, kG), 256, 0, stream>>>(x, mask, out);
  } else {
    dropblock_fused<<<dim3(kC, kG), 256, 0, stream>>>(x, noise, out);
  }

  (void)in_sizes; (void)n_in; (void)out_size;
}